// GCAE_38809324486725
// MI455X (gfx1250) — compile-verified
//
#include <hip/hip_runtime.h>
#include <hip/hip_bf16.h>

typedef float v2f __attribute__((ext_vector_type(2)));
typedef float v8f __attribute__((ext_vector_type(8)));
typedef unsigned int u32x4 __attribute__((ext_vector_type(4)));
typedef int i32x4 __attribute__((ext_vector_type(4)));
typedef int i32x8 __attribute__((ext_vector_type(8)));

#define HEADS 4
#define CH    32
#define HC    128
#define NEG_SLOPE 0.2f

// padded LDS strides (bank-conflict-free for half-wave K-split fragment reads)
#define W1S 136   // We1: 128 + 8 pad  -> half-wave bank sets disjoint
#define W2S 40    // We2: 32 + 8 pad
#define H1S 132   // h1 : 128 + 4 pad  -> row*4 bank stepping, conflict-free

#define ENC_WAVES 8   // 256-thread encoder blocks: 8 waves x 16 rows = 128 rows/block

// ---------------------------------------------------------------- utilities
__device__ __forceinline__ void atomicMaxF(float* addr, float val) {
    if (val >= 0.f) {
        atomicMax((int*)addr, __float_as_int(val));
    } else {
        atomicMin((unsigned int*)addr, __float_as_uint(val));
    }
}

__global__ void fill_f32_kernel(float* __restrict__ p, float v, long long n) {
    long long i = (long long)blockIdx.x * blockDim.x + threadIdx.x;
    if (i < n) p[i] = v;
}

// -------------------------------------------- TDM: 2D fp32 tile -> LDS w/ row pad
// dims: dim0 = row length (elems), dim1 = #rows, contiguous (stride0 = dim0)
// padI/padA: D# pad_interval / pad_amount codes (pad applied to LDS dest)
__device__ __forceinline__ void tdm_load_2d_f32(const void* gptr, unsigned ldsOff,
                                                unsigned dim0, unsigned dim1,
                                                unsigned padI, unsigned padA) {
    unsigned long long ga = (unsigned long long)gptr;
    u32x4 g0;
    g0[0] = 1u;                                     // count=1 valid user descriptor
    g0[1] = ldsOff;                                 // lds_addr (bytes)
    g0[2] = (unsigned)(ga & 0xFFFFFFFFull);         // global_addr[31:0]
    g0[3] = (unsigned)((ga >> 32) & 0x01FFFFFFull)  // global_addr[56:32]
          | 0x80000000u;                            // type = 2 ("image")
    i32x8 g1;
    g1[0] = (int)((2u << 16)                        // data_size = 4B
          | (1u << 20)                              // pad_enable
          | (padI << 22) | (padA << 25));           // pad config
    g1[1] = (int)((dim0 & 0xFFFFu) << 16);          // tensor_dim0 lo16
    g1[2] = (int)((dim0 >> 16) | ((dim1 & 0xFFFFu) << 16));   // dim0 hi / dim1 lo
    g1[3] = (int)((dim1 >> 16) | ((dim0 & 0xFFFFu) << 16));   // dim1 hi / tile_dim0
    g1[4] = (int)(dim1 & 0xFFFFu);                  // tile_dim1 (tile_dim2 = 0)
    g1[5] = (int)dim0;                              // tensor_dim0_stride lo32
    g1[6] = 0;
    g1[7] = 0;
    i32x4 z4 = {0, 0, 0, 0};
#if __clang_major__ >= 23
    i32x8 z8 = {0, 0, 0, 0, 0, 0, 0, 0};
    __builtin_amdgcn_tensor_load_to_lds(g0, g1, z4, z4, z8, 0);
#else
    __builtin_amdgcn_tensor_load_to_lds(g0, g1, z4, z4, 0);
#endif
}

// ------------------------------------------------- fold attention projections
__global__ void precompute_vecs_kernel(const float* __restrict__ Wg,
                                       const float* __restrict__ as,
                                       const float* __restrict__ ad,
                                       const float* __restrict__ We,
                                       const float* __restrict__ ae,
                                       float* __restrict__ vecs) {
    int t = threadIdx.x;
    if (t < 12) {
        int d = t >> 2, h = t & 3;
        float s1 = 0.f, s2 = 0.f;
        for (int c = 0; c < CH; ++c) {
            float wv = Wg[d * HC + h * CH + c];
            s1 += wv * as[h * CH + c];
            s2 += wv * ad[h * CH + c];
        }
        vecs[t] = s1; vecs[12 + t] = s2;
    }
    if (t < 28) {
        int d = t >> 2, h = t & 3;
        float s = 0.f;
        for (int c = 0; c < CH; ++c)
            s += We[d * HC + h * CH + c] * ae[h * CH + c];
        vecs[24 + t] = s;
    }
}

// ------------------------------------------------------- node projection (K=3)
__global__ void node_xl_kernel(const float* __restrict__ x,
                               const float* __restrict__ Wg,
                               float* __restrict__ xl, long long total) {
    long long idx = (long long)blockIdx.x * blockDim.x + threadIdx.x;
    if (idx >= total) return;
    long long nn = idx >> 7;
    int j = (int)(idx & 127);
    const float* xp = x + nn * 3;
    xl[idx] = xp[0] * Wg[j] + xp[1] * Wg[HC + j] + xp[2] * Wg[2 * HC + j];
}

__global__ void node_scalars_kernel(const float* __restrict__ x,
                                    const float* __restrict__ vecs,
                                    float* __restrict__ asrc,
                                    float* __restrict__ adst, int n) {
    int nn = blockIdx.x * blockDim.x + threadIdx.x;
    if (nn >= n) return;
    float x0 = x[nn * 3], x1 = x[nn * 3 + 1], x2 = x[nn * 3 + 2];
    #pragma unroll
    for (int h = 0; h < HEADS; ++h) {
        asrc[nn * 4 + h] = x0 * vecs[h] + x1 * vecs[4 + h] + x2 * vecs[8 + h];
        adst[nn * 4 + h] = x0 * vecs[12 + h] + x1 * vecs[16 + h] + x2 * vecs[20 + h];
    }
}

// ----------------------------------------------------------- edge pass 1
__global__ void edge_phase1_kernel(const int* __restrict__ ei,
                                   const float* __restrict__ eattr,
                                   const float* __restrict__ asrc,
                                   const float* __restrict__ adst,
                                   const float* __restrict__ vecs,
                                   float* __restrict__ alpha,
                                   float* __restrict__ amax,
                                   float* __restrict__ aesum,
                                   float* __restrict__ cntv, int e) {
    int idx = blockIdx.x * blockDim.x + threadIdx.x;
    if (idx >= e) return;
    int s = ei[idx], d = ei[e + idx];
    float at[7];
    #pragma unroll
    for (int i = 0; i < 7; ++i) at[i] = eattr[(long long)idx * 7 + i];
    atomicAdd(&cntv[d], 1.f);
    #pragma unroll
    for (int h = 0; h < HEADS; ++h) {
        float ae = 0.f;
        #pragma unroll
        for (int i = 0; i < 7; ++i) ae += at[i] * vecs[24 + i * 4 + h];
        float al = asrc[s * 4 + h] + adst[d * 4 + h] + ae;
        al = al > 0.f ? al : NEG_SLOPE * al;
        alpha[(long long)idx * 4 + h] = al;
        atomicMaxF(&amax[d * 4 + h], al);
        atomicAdd(&aesum[d * 4 + h], ae);
    }
}

// ----------------------------------------------------------- self-loop pass 1
__global__ void selfloop1_kernel(const float* __restrict__ asrc,
                                 const float* __restrict__ adst,
                                 const float* __restrict__ aesum,
                                 const float* __restrict__ cntv,
                                 float* __restrict__ aloop,
                                 float* __restrict__ amax, int n) {
    int nn = blockIdx.x * blockDim.x + threadIdx.x;
    if (nn >= n) return;
    float c = cntv[nn];
    float dv = c > 1.f ? c : 1.f;
    #pragma unroll
    for (int h = 0; h < HEADS; ++h) {
        float ael = aesum[nn * 4 + h] / dv;
        float al = asrc[nn * 4 + h] + adst[nn * 4 + h] + ael;
        al = al > 0.f ? al : NEG_SLOPE * al;
        aloop[nn * 4 + h] = al;
        float m = amax[nn * 4 + h];
        amax[nn * 4 + h] = al > m ? al : m;
    }
}

// ----------------------------------------------------------- edge pass 2 (exp+sum)
__global__ void edge_phase2_kernel(const int* __restrict__ ei,
                                   float* __restrict__ alpha,
                                   const float* __restrict__ amax,
                                   float* __restrict__ denom, int e) {
    int idx = blockIdx.x * blockDim.x + threadIdx.x;
    if (idx >= e) return;
    int d = ei[e + idx];
    #pragma unroll
    for (int h = 0; h < HEADS; ++h) {
        float ex = __expf(alpha[(long long)idx * 4 + h] - amax[d * 4 + h]);
        alpha[(long long)idx * 4 + h] = ex;
        atomicAdd(&denom[d * 4 + h], ex);
    }
}

// ------------------------------------------- self-loop pass 2 (finalize denom, weight)
__global__ void selfloop2_kernel(float* __restrict__ aloop,
                                 const float* __restrict__ amax,
                                 float* __restrict__ denom, int n) {
    int nn = blockIdx.x * blockDim.x + threadIdx.x;
    if (nn >= n) return;
    #pragma unroll
    for (int h = 0; h < HEADS; ++h) {
        float exl = __expf(aloop[nn * 4 + h] - amax[nn * 4 + h]);
        float df = denom[nn * 4 + h] + exl;
        denom[nn * 4 + h] = df;
        aloop[nn * 4 + h] = exl / (df + 1e-16f);
    }
}

// -------------------------------- init GAT output with self-loop message (no atomics)
__global__ void init_out_kernel(const float* __restrict__ aloop,
                                const float* __restrict__ xl,
                                float* __restrict__ gatout, long long total) {
    long long idx = (long long)blockIdx.x * blockDim.x + threadIdx.x;
    if (idx >= total) return;
    long long nn = idx >> 7;
    int j = (int)(idx & 127);
    int h = j >> 5;
    gatout[idx] = aloop[nn * 4 + h] * xl[idx];
}

// --------------------------------------- edge messages: one wave (32 lanes) per edge
__global__ __launch_bounds__(256) void edge_msg_kernel(const int* __restrict__ ei,
                                                       const float* __restrict__ alpha,
                                                       const float* __restrict__ denom,
                                                       const float* __restrict__ xl,
                                                       float* __restrict__ gatout, int e) {
    int wid = blockIdx.x * (blockDim.x >> 5) + (threadIdx.x >> 5);
    int lane = threadIdx.x & 31;
    if (wid >= e) return;
    int s = ei[wid], d = ei[e + wid];
    int h = lane >> 3;
    float wgt = alpha[(long long)wid * 4 + h] / (denom[d * 4 + h] + 1e-16f);
    const float4* x4 = (const float4*)(xl + (long long)s * HC);
    float4 v = x4[lane];
    float* o = gatout + (long long)d * HC + lane * 4;
    atomicAdd(o + 0, wgt * v.x);
    atomicAdd(o + 1, wgt * v.y);
    atomicAdd(o + 2, wgt * v.z);
    atomicAdd(o + 3, wgt * v.w);
}

// -------------------------------------------------------------- fused encoder
// h = elu(gat + b_gat); h1 = prelu(h@We1 + be1); z = h1@We2 + be2
// Weights staged LDS-resident via TDM (row-padded against bank conflicts);
// one wave per 16-node tile; fp32 WMMA 16x16x4 keeps exact reference precision.
// n0 loops held at unroll 1: single live accumulator -> <128 VGPRs, no MSB thrash.
__global__ __launch_bounds__(256) void encoder_wmma_kernel(
        const float* __restrict__ gat, const float* __restrict__ bg,
        const float* __restrict__ We1, const float* __restrict__ be1,
        const float* __restrict__ pa,  const float* __restrict__ We2,
        const float* __restrict__ be2, float* __restrict__ z, int n) {
    __shared__ float lWe1[HC * W1S];                 // 69,632 B
    __shared__ float lWe2[HC * W2S];                 // 20,480 B
    __shared__ float h1s[ENC_WAVES * 16 * H1S];      // 67,584 B re-stripe scratch

    // stage weights once per block via Tensor Data Mover
    if (threadIdx.x == 0) {
        // We1: 128 rows x 128 cols; +8 DWORD LDS pad per 128 DWORDs (codes 6,7)
        tdm_load_2d_f32(We1, (unsigned)(unsigned long long)(void*)lWe1, HC, HC, 6u, 7u);
        // We2: 128 rows x 32 cols;  +8 DWORD LDS pad per 32 DWORDs  (codes 4,7)
        tdm_load_2d_f32(We2, (unsigned)(unsigned long long)(void*)lWe2, 32u, HC, 4u, 7u);
        __builtin_amdgcn_s_wait_tensorcnt(0);
    }
    __syncthreads();

    const int wave = threadIdx.x >> 5;
    const int lane = threadIdx.x & 31;
    const int row  = lane & 15;
    const int hi   = lane >> 4;
    const int koff = hi * 2;
    const int base = (blockIdx.x * ENC_WAVES + wave) * 16;

    if (base < n) {
        const float a_p = pa[0];

        // A fragments: whole 16x128 tile in 64 VGPRs, fused bias+ELU on load
        v2f afrag[32];
        #pragma unroll
        for (int kk = 0; kk < 32; ++kk) {
            const float* p = gat + (long long)(base + row) * HC + kk * 4 + koff;
            float x0 = p[0] + bg[kk * 4 + koff];
            float x1 = p[1] + bg[kk * 4 + koff + 1];
            x0 = x0 >= 0.f ? x0 : (__expf(x0) - 1.f);
            x1 = x1 >= 0.f ? x1 : (__expf(x1) - 1.f);
            afrag[kk].x = x0; afrag[kk].y = x1;
        }

        float* myh1 = h1s + wave * (16 * H1S);

        // GEMM1: 16x128 @ 128x128 (8 column tiles x 32 K-steps), B from LDS
        #pragma unroll 1
        for (int n0 = 0; n0 < 8; ++n0) {
            const int col = n0 * 16 + row;
            float bv = be1[col];
            v8f acc = {bv, bv, bv, bv, bv, bv, bv, bv};
            #pragma unroll
            for (int kk = 0; kk < 32; ++kk) {
                int k = kk * 4 + koff;
                v2f b;
                b.x = lWe1[k * W1S + col];
                b.y = lWe1[(k + 1) * W1S + col];
                acc = __builtin_amdgcn_wmma_f32_16x16x4_f32(
                    false, afrag[kk], false, b, (short)0, acc, false, false);
            }
            #pragma unroll
            for (int r = 0; r < 8; ++r) {      // PReLU + D->A re-stripe via LDS
                float v = acc[r];
                v = v >= 0.f ? v : a_p * v;
                myh1[(r + hi * 8) * H1S + col] = v;
            }
        }

        // GEMM2: 16x128 @ 128x32 (2 column tiles x 32 K-steps)
        #pragma unroll 1
        for (int n0 = 0; n0 < 2; ++n0) {
            const int col = n0 * 16 + row;
            float bv = be2[col];
            v8f acc = {bv, bv, bv, bv, bv, bv, bv, bv};
            #pragma unroll
            for (int kk = 0; kk < 32; ++kk) {
                int k = kk * 4 + koff;
                v2f a2, b;
                a2.x = myh1[row * H1S + k];
                a2.y = myh1[row * H1S + k + 1];
                b.x = lWe2[k * W2S + col];
                b.y = lWe2[(k + 1) * W2S + col];
                acc = __builtin_amdgcn_wmma_f32_16x16x4_f32(
                    false, a2, false, b, (short)0, acc, false, false);
            }
            #pragma unroll
            for (int r = 0; r < 8; ++r)
                z[(long long)(base + r + hi * 8) * 32 + col] = acc[r];
        }
    }
}

// -------------------------------------------------------------------- pooling
__global__ void pool_kernel(const float* __restrict__ z,
                            const int* __restrict__ batch,
                            float* __restrict__ psum, float* __restrict__ pcnt,
                            long long total) {
    long long idx = (long long)blockIdx.x * blockDim.x + threadIdx.x;
    if (idx >= total) return;
    long long nn = idx >> 5;
    int c = (int)(idx & 31);
    int b = batch[nn];
    atomicAdd(&psum[b * 32 + c], z[idx]);
    if (c == 0) atomicAdd(&pcnt[b], 1.f);
}

// ------------------------------------------------------------------ predictor
__global__ void predictor_kernel(const float* __restrict__ psum,
                                 const float* __restrict__ pcnt,
                                 const float* __restrict__ Wp1, const float* __restrict__ bp1,
                                 const float* __restrict__ Wp2, const float* __restrict__ bp2,
                                 const float* __restrict__ Wp3, const float* __restrict__ bp3,
                                 float* __restrict__ pred, int ngraphs) {
    int g = blockIdx.x * blockDim.x + threadIdx.x;
    if (g >= ngraphs) return;
    float c = pcnt[g]; c = c > 1.f ? c : 1.f;
    float m[32];
    for (int i = 0; i < 32; ++i) m[i] = psum[g * 32 + i] / c;
    float p1[128];
    for (int j = 0; j < 128; ++j) {
        float s = bp1[j];
        for (int i = 0; i < 32; ++i) s += m[i] * Wp1[i * 128 + j];
        p1[j] = s / (1.f + __expf(-s));
    }
    float p2[64];
    for (int j = 0; j < 64; ++j) {
        float s = bp2[j];
        for (int i = 0; i < 128; ++i) s += p1[i] * Wp2[i * 64 + j];
        p2[j] = s / (1.f + __expf(-s));
    }
    float s = bp3[0];
    for (int i = 0; i < 64; ++i) s += p2[i] * Wp3[i];
    pred[g] = s;
}

// ================================================================== launcher
extern "C" void kernel_launch(void* const* d_in, const int* in_sizes, int n_in,
                              void* d_out, int out_size, void* d_ws, size_t ws_size,
                              hipStream_t stream) {
    const float* x         = (const float*)d_in[0];
    const float* edge_attr = (const float*)d_in[1];
    const int*   edge_idx  = (const int*)d_in[2];
    const int*   batch     = (const int*)d_in[3];
    const float* W_gat     = (const float*)d_in[4];
    const float* att_src   = (const float*)d_in[5];
    const float* att_dst   = (const float*)d_in[6];
    const float* att_edge  = (const float*)d_in[7];
    const float* W_edge    = (const float*)d_in[8];
    const float* b_gat     = (const float*)d_in[9];
    const float* W_e1      = (const float*)d_in[10];
    const float* b_e1      = (const float*)d_in[11];
    const float* prelu_a   = (const float*)d_in[12];
    const float* W_e2      = (const float*)d_in[13];
    const float* b_e2      = (const float*)d_in[14];
    const float* W_p1      = (const float*)d_in[15];
    const float* b_p1      = (const float*)d_in[16];
    const float* W_p2      = (const float*)d_in[17];
    const float* b_p2      = (const float*)d_in[18];
    const float* W_p3      = (const float*)d_in[19];
    const float* b_p3      = (const float*)d_in[20];

    const int n = in_sizes[0] / 3;
    const int e = in_sizes[1] / 7;
    const int ngraphs = out_size - n * 32;

    float* pred = (float*)d_out;
    float* z    = pred + ngraphs;

    float* w = (float*)d_ws;
    float* xl     = w; w += (long long)n * HC;
    float* asrc   = w; w += (long long)n * 4;
    float* adst   = w; w += (long long)n * 4;
    float* aesum  = w; w += (long long)n * 4;
    float* cntv   = w; w += n;
    float* amax   = w; w += (long long)n * 4;
    float* denom  = w; w += (long long)n * 4;
    float* alpha  = w; w += (long long)e * 4;
    float* aloop  = w; w += (long long)n * 4;
    float* gatout = w; w += (long long)n * HC;
    float* psum   = w; w += (long long)ngraphs * 32;
    float* pcnt   = w; w += ngraphs;
    float* vecs   = w; w += 64;

    hipMemsetAsync(aesum, 0, (size_t)n * 4 * sizeof(float), stream);
    hipMemsetAsync(cntv,  0, (size_t)n * sizeof(float), stream);
    hipMemsetAsync(denom, 0, (size_t)n * 4 * sizeof(float), stream);
    hipMemsetAsync(psum,  0, (size_t)ngraphs * 32 * sizeof(float), stream);
    hipMemsetAsync(pcnt,  0, (size_t)ngraphs * sizeof(float), stream);
    {
        long long t = (long long)n * 4;
        fill_f32_kernel<<<(int)((t + 255) / 256), 256, 0, stream>>>(amax, -INFINITY, t);
    }

    precompute_vecs_kernel<<<1, 32, 0, stream>>>(W_gat, att_src, att_dst, W_edge, att_edge, vecs);

    {
        long long t = (long long)n * HC;
        node_xl_kernel<<<(int)((t + 255) / 256), 256, 0, stream>>>(x, W_gat, xl, t);
    }
    node_scalars_kernel<<<(n + 255) / 256, 256, 0, stream>>>(x, vecs, asrc, adst, n);

    edge_phase1_kernel<<<(e + 255) / 256, 256, 0, stream>>>(edge_idx, edge_attr, asrc, adst,
                                                            vecs, alpha, amax, aesum, cntv, e);
    selfloop1_kernel<<<(n + 255) / 256, 256, 0, stream>>>(asrc, adst, aesum, cntv, aloop, amax, n);
    edge_phase2_kernel<<<(e + 255) / 256, 256, 0, stream>>>(edge_idx, alpha, amax, denom, e);
    selfloop2_kernel<<<(n + 255) / 256, 256, 0, stream>>>(aloop, amax, denom, n);
    {
        long long t = (long long)n * HC;
        init_out_kernel<<<(int)((t + 255) / 256), 256, 0, stream>>>(aloop, xl, gatout, t);
    }
    edge_msg_kernel<<<(e + 7) / 8, 256, 0, stream>>>(edge_idx, alpha, denom, xl, gatout, e);

    encoder_wmma_kernel<<<(n + 16 * ENC_WAVES - 1) / (16 * ENC_WAVES), 32 * ENC_WAVES, 0, stream>>>(
        gatout, b_gat, W_e1, b_e1, prelu_a, W_e2, b_e2, z, n);

    {
        long long t = (long long)n * 32;
        pool_kernel<<<(int)((t + 255) / 256), 256, 0, stream>>>(z, batch, psum, pcnt, t);
    }
    predictor_kernel<<<1, 64, 0, stream>>>(psum, pcnt, W_p1, b_p1, W_p2, b_p2, W_p3, b_p3,
                                           pred, ngraphs);
}